// ManualGATLayer_50113678410370
// MI455X (gfx1250) — compile-verified
//
#include <hip/hip_runtime.h>
#include <hip/hip_bf16.h>
#include <math.h>

#define HEADS 4
#define DHEAD 32
#define ODIM 128
#define EEDIM 16

typedef __attribute__((ext_vector_type(2))) float v2f;
typedef __attribute__((ext_vector_type(8))) float v8f;

// ---------------------------------------------------------------------------
// Kernel 1: h = x @ W  via V_WMMA_F32_16X16X4_F32.
// Block = 256 threads = 8 waves. Each wave computes one 16x16 tile of h:
// rows [blockIdx.x*16, +16), cols [wave*16, +16). K=128 -> 32 WMMA(K=4) ops.
// ---------------------------------------------------------------------------
__global__ void k_gemm(const float* __restrict__ x, const float* __restrict__ W,
                       float* __restrict__ h, int N) {
  const int wave  = threadIdx.x >> 5;
  const int lane  = threadIdx.x & 31;
  const int lr    = lane & 15;          // row-in-tile (A) / col-in-tile (B,C)
  const int khalf = lane >> 4;          // 0: K=0,1   1: K=2,3
  const int m0 = blockIdx.x * 16;
  const int n0 = wave * 16;

  // clamp row for loads on a (non-existent here: 50000 = 3125*16) tail tile
  int arow = m0 + lr; if (arow >= N) arow = N - 1;
  const float* __restrict__ xrow = x + (long)arow * ODIM;

  v8f c = {0.f, 0.f, 0.f, 0.f, 0.f, 0.f, 0.f, 0.f};

#pragma unroll
  for (int k = 0; k < ODIM; k += 4) {
    const int kk = k + khalf * 2;
    v2f a, b;
    a.x = xrow[kk];
    a.y = xrow[kk + 1];
    b.x = W[(long)kk * ODIM + n0 + lr];
    b.y = W[(long)(kk + 1) * ODIM + n0 + lr];
    // 8 args: (neg_a, A, neg_b, B, c_mod, C, reuse_a, reuse_b)
    c = __builtin_amdgcn_wmma_f32_16x16x4_f32(false, a, false, b,
                                              (short)0, c, false, false);
  }

  // C/D layout: VGPR r, lanes 0-15 -> (M=r, N=lane); lanes 16-31 -> (M=r+8)
  const int mbase = m0 + khalf * 8;
  const int col   = n0 + lr;
#pragma unroll
  for (int r = 0; r < 8; ++r) {
    const int row = mbase + r;
    if (row < N) h[(long)row * ODIM + col] = c[r];
  }
}

// ---------------------------------------------------------------------------
// Kernel 2: init out_acc=0, amax=-1e9, asum=0, and the 3x4 edge-type table
//           table[t][hh] = sum_j edge_embed[t][j] * a_edge[hh][j]
// ---------------------------------------------------------------------------
__global__ void k_init(float* __restrict__ out_acc, float* __restrict__ amax,
                       float* __restrict__ asum, float* __restrict__ table,
                       const float* __restrict__ edge_embed,
                       const float* __restrict__ a_edge, int N) {
  const long gid = (long)blockIdx.x * blockDim.x + threadIdx.x;
  if (gid < (long)N * ODIM) out_acc[gid] = 0.f;
  if (gid < (long)N * HEADS) { amax[gid] = -1e9f; asum[gid] = 0.f; }
  if (gid < 3 * HEADS) {
    const int t = (int)(gid >> 2), hh = (int)(gid & 3);
    float s = 0.f;
#pragma unroll
    for (int j = 0; j < EEDIM; ++j)
      s += edge_embed[t * EEDIM + j] * a_edge[hh * EEDIM + j];
    table[gid] = s;
  }
}

// ---------------------------------------------------------------------------
// Kernel 3: per-node attention scores  s_src[n,h], s_dst[n,h]
// ---------------------------------------------------------------------------
__global__ void k_scores(const float* __restrict__ h,
                         const float* __restrict__ a_src,
                         const float* __restrict__ a_dst,
                         float* __restrict__ s_src, float* __restrict__ s_dst,
                         int N) {
  const long gid = (long)blockIdx.x * blockDim.x + threadIdx.x;
  if (gid >= (long)N * HEADS) return;
  const int n = (int)(gid >> 2), hh = (int)(gid & 3);
  const float* __restrict__ hp = h + (long)n * ODIM + hh * DHEAD;
  float ss = 0.f, sd = 0.f;
#pragma unroll
  for (int d = 0; d < DHEAD; ++d) {
    const float v = hp[d];
    ss += v * a_src[hh * DHEAD + d];
    sd += v * a_dst[hh * DHEAD + d];
  }
  s_src[gid] = ss;
  s_dst[gid] = sd;
}

__device__ __forceinline__ float edge_logit(const float* s_src, const float* s_dst,
                                            const float* table, int src, int dst,
                                            int et, int hh) {
  float a = s_src[src * HEADS + hh] + s_dst[dst * HEADS + hh] + table[et * HEADS + hh];
  return a > 0.f ? a : 0.2f * a;  // LeakyReLU(0.2)
}

// ---------------------------------------------------------------------------
// Kernel 4: segment max of logits over dst  (atomic float max)
// ---------------------------------------------------------------------------
__global__ void k_edge_max(const int* __restrict__ ei, const int* __restrict__ et,
                           const float* __restrict__ s_src,
                           const float* __restrict__ s_dst,
                           const float* __restrict__ table,
                           float* __restrict__ amax, long E) {
  const long gid = (long)blockIdx.x * blockDim.x + threadIdx.x;
  if (gid >= E * HEADS) return;
  const long e = gid >> 2;
  const int hh = (int)(gid & 3);
  const int src = ei[e], dst = ei[E + e];
  const float a = edge_logit(s_src, s_dst, table, src, dst, et[e], hh);
  atomicMax(&amax[dst * HEADS + hh], a);
}

// ---------------------------------------------------------------------------
// Kernel 5: attn_exp = exp(logit - amax[dst]); segment-sum into asum
// ---------------------------------------------------------------------------
__global__ void k_edge_exp(const int* __restrict__ ei, const int* __restrict__ et,
                           const float* __restrict__ s_src,
                           const float* __restrict__ s_dst,
                           const float* __restrict__ table,
                           const float* __restrict__ amax,
                           float* __restrict__ attn_exp,
                           float* __restrict__ asum, long E) {
  const long gid = (long)blockIdx.x * blockDim.x + threadIdx.x;
  if (gid >= E * HEADS) return;
  const long e = gid >> 2;
  const int hh = (int)(gid & 3);
  const int src = ei[e], dst = ei[E + e];
  const float a  = edge_logit(s_src, s_dst, table, src, dst, et[e], hh);
  const float ex = __expf(a - amax[dst * HEADS + hh]);
  attn_exp[gid] = ex;
  atomicAdd(&asum[dst * HEADS + hh], ex);
}

// ---------------------------------------------------------------------------
// Kernel 6: weighted scatter  out_acc[dst, c] += w(e,h) * h[src, c]
// one thread per (edge, column); columns coalesced, atomics land in L2.
// ---------------------------------------------------------------------------
__global__ void k_scatter(const int* __restrict__ ei,
                          const float* __restrict__ h,
                          const float* __restrict__ attn_exp,
                          const float* __restrict__ asum,
                          float* __restrict__ out_acc, long E) {
  const long gid = (long)blockIdx.x * blockDim.x + threadIdx.x;
  if (gid >= E * ODIM) return;
  const long e = gid >> 7;
  const int c  = (int)(gid & (ODIM - 1));
  const int hh = c >> 5;
  const int src = ei[e], dst = ei[E + e];
  const float w = attn_exp[e * HEADS + hh] / (asum[dst * HEADS + hh] + 1e-10f);
  atomicAdd(&out_acc[(long)dst * ODIM + c], w * h[(long)src * ODIM + c]);
}

// ---------------------------------------------------------------------------
// Kernel 7: residual + LayerNorm. One wave32 per node, 4 cols per lane.
// ---------------------------------------------------------------------------
__global__ void k_lnorm(const float* __restrict__ out_acc,
                        const float* __restrict__ h,
                        const float* __restrict__ gamma,
                        const float* __restrict__ beta,
                        float* __restrict__ out, int N) {
  const int wave = threadIdx.x >> 5, lane = threadIdx.x & 31;
  const int n = blockIdx.x * (blockDim.x >> 5) + wave;
  if (n >= N) return;
  float v[4];
  float s = 0.f;
#pragma unroll
  for (int i = 0; i < 4; ++i) {
    const int c = lane + i * 32;
    v[i] = out_acc[(long)n * ODIM + c] + h[(long)n * ODIM + c];
    s += v[i];
  }
#pragma unroll
  for (int off = 16; off > 0; off >>= 1) s += __shfl_xor(s, off, 32);
  const float mean = s * (1.f / ODIM);
  float q = 0.f;
#pragma unroll
  for (int i = 0; i < 4; ++i) { const float d = v[i] - mean; q += d * d; }
#pragma unroll
  for (int off = 16; off > 0; off >>= 1) q += __shfl_xor(q, off, 32);
  const float rstd = rsqrtf(q * (1.f / ODIM) + 1e-5f);
#pragma unroll
  for (int i = 0; i < 4; ++i) {
    const int c = lane + i * 32;
    out[(long)n * ODIM + c] = (v[i] - mean) * rstd * gamma[c] + beta[c];
  }
}

// ---------------------------------------------------------------------------
extern "C" void kernel_launch(void* const* d_in, const int* in_sizes, int n_in,
                              void* d_out, int out_size, void* d_ws, size_t ws_size,
                              hipStream_t stream) {
  const float* x          = (const float*)d_in[0];
  const float* W          = (const float*)d_in[1];
  const float* a_src      = (const float*)d_in[2];
  const float* a_dst      = (const float*)d_in[3];
  const float* a_edge     = (const float*)d_in[4];
  const float* edge_embed = (const float*)d_in[5];
  const float* ln_gamma   = (const float*)d_in[6];
  const float* ln_beta    = (const float*)d_in[7];
  const int*   edge_index = (const int*)d_in[8];
  const int*   edge_type  = (const int*)d_in[9];
  float* out = (float*)d_out;

  const int  N = in_sizes[0] / ODIM;
  const long E = (long)in_sizes[8] / 2;

  // workspace layout (floats)
  float* ws       = (float*)d_ws;
  float* h        = ws;                       // N*128
  float* out_acc  = h       + (long)N * ODIM; // N*128
  float* s_src    = out_acc + (long)N * ODIM; // N*4
  float* s_dst    = s_src   + (long)N * HEADS;
  float* amax     = s_dst   + (long)N * HEADS;
  float* asum     = amax    + (long)N * HEADS;
  float* attn_exp = asum    + (long)N * HEADS; // E*4
  float* table    = attn_exp + E * HEADS;      // 12 (+pad)

  // 1) GEMM h = x @ W  (WMMA f32)
  {
    dim3 grid((N + 15) / 16), block(256);
    k_gemm<<<grid, block, 0, stream>>>(x, W, h, N);
  }
  // 2) init accumulators + edge-type table
  {
    const long n = (long)N * ODIM;
    k_init<<<(unsigned)((n + 255) / 256), 256, 0, stream>>>(
        out_acc, amax, asum, table, edge_embed, a_edge, N);
  }
  // 3) per-node scores
  {
    const long n = (long)N * HEADS;
    k_scores<<<(unsigned)((n + 255) / 256), 256, 0, stream>>>(
        h, a_src, a_dst, s_src, s_dst, N);
  }
  // 4) segment max
  {
    const long n = E * HEADS;
    k_edge_max<<<(unsigned)((n + 255) / 256), 256, 0, stream>>>(
        edge_index, edge_type, s_src, s_dst, table, amax, E);
  }
  // 5) exp + segment sum
  {
    const long n = E * HEADS;
    k_edge_exp<<<(unsigned)((n + 255) / 256), 256, 0, stream>>>(
        edge_index, edge_type, s_src, s_dst, table, amax, attn_exp, asum, E);
  }
  // 6) weighted scatter
  {
    const long n = E * ODIM;
    k_scatter<<<(unsigned)((n + 255) / 256), 256, 0, stream>>>(
        edge_index, h, attn_exp, asum, out_acc, E);
  }
  // 7) residual + LayerNorm
  {
    dim3 block(128);  // 4 waves -> 4 nodes per block
    dim3 grid((N + 3) / 4);
    k_lnorm<<<grid, block, 0, stream>>>(out_acc, h, ln_gamma, ln_beta, out, N);
  }
}